// Model_65618510348712
// MI455X (gfx1250) — compile-verified
//
#include <hip/hip_runtime.h>
#include <math.h>

// Problem constants (match reference)
#define BB   4
#define NN   4096
#define CC   256
#define HH   8
#define HD   32
#define MM   1366      // ceil(4096/3) strided keys
#define MP   1376      // padded to 86*16
#define MT   86        // 16-wide key tiles
#define KT   344       // MP/4 : K-steps for AV (f32 WMMA has K=4)
#define HID  512
#define EPSL 1e-6f

typedef float v2f  __attribute__((ext_vector_type(2)));
typedef float v8f  __attribute__((ext_vector_type(8)));
typedef unsigned int u32x4 __attribute__((ext_vector_type(4)));
typedef int   i32x4 __attribute__((ext_vector_type(4)));
typedef int   i32x8 __attribute__((ext_vector_type(8)));

// D = A(16x4 f32) * B(4x16 f32) + C(16x16 f32)
__device__ __forceinline__ v8f wmma_f32(v2f a, v2f b, v8f c) {
  return __builtin_amdgcn_wmma_f32_16x16x4_f32(false, a, false, b, (short)0, c,
                                               false, false);
}

// TDM: async-load a [rows_valid x 32] fp32 tile (row stride 3*CC elements)
// into LDS at lds_addr. Rows >= rows_valid read as zero (hardware OOB).
// D# layout per CDNA5 ISA 8.3/8.4: count=1, type=2, data_size=4B,
// tile_dim0=32, tile_dim1=16, tensor_dim0=32, tensor_dim1=rows_valid,
// tensor_dim0_stride=3*CC.  This toolchain's builtin takes 6 args
// (g0, g1, g2, g3, g4, cpol); groups 2-4 unused for a 2D tensor.
__device__ __forceinline__ void tdm_load_kv_tile(unsigned lds_addr,
                                                 const float* gptr,
                                                 int rows_valid) {
  unsigned long long ga = (unsigned long long)(uintptr_t)gptr;
  u32x4 g0;
  g0[0] = 1u;                                                  // count=1
  g0[1] = lds_addr;                                            // LDS dest
  g0[2] = (unsigned)ga;                                        // gaddr lo
  g0[3] = ((unsigned)(ga >> 32) & 0x1FFFFFFu) | 0x80000000u;   // hi | type=2
  i32x8 g1;
  g1[0] = (int)(2u << 16);                    // data_size = 4 bytes
  g1[1] = (int)(32u << 16);                   // tensor_dim0 = 32 (bits 63:48)
  g1[2] = (int)((unsigned)rows_valid << 16);  // tensor_dim1 lo (bits 95:80)
  g1[3] = (int)(32u << 16);                   // tile_dim0 = 32 (bits 127:112)
  g1[4] = 16;                                 // tile_dim1 = 16 (bits 143:128)
  g1[5] = 3 * CC;                             // tensor_dim0_stride = 768
  g1[6] = 0;
  g1[7] = 0;
  i32x4 z4 = {0, 0, 0, 0};
  i32x8 z8 = {0, 0, 0, 0, 0, 0, 0, 0};
  __builtin_amdgcn_tensor_load_to_lds(g0, g1, z4, z4, z8, 0);
}

// ---------------------------------------------------------------- LayerNorm1
__global__ __launch_bounds__(256) void k_ln1(const float* __restrict__ x,
                                             const float* __restrict__ g,
                                             const float* __restrict__ b,
                                             float* __restrict__ y) {
  int wave = threadIdx.x >> 5;
  int lane = threadIdx.x & 31;
  size_t row = (size_t)blockIdx.x * 8 + wave;          // [0, B*N)
  const float* xr = x + row * CC;
  float4 v0 = *(const float4*)(xr + lane * 4);
  float4 v1 = *(const float4*)(xr + 128 + lane * 4);
  float s = v0.x + v0.y + v0.z + v0.w + v1.x + v1.y + v1.z + v1.w;
  float q = v0.x * v0.x + v0.y * v0.y + v0.z * v0.z + v0.w * v0.w +
            v1.x * v1.x + v1.y * v1.y + v1.z * v1.z + v1.w * v1.w;
#pragma unroll
  for (int m = 16; m; m >>= 1) {
    s += __shfl_xor(s, m);
    q += __shfl_xor(q, m);
  }
  float mean = s * (1.0f / CC);
  float var = q * (1.0f / CC) - mean * mean;
  float inv = rsqrtf(var + EPSL);
  float* yr = y + row * CC;
  int c0 = lane * 4, c1 = 128 + lane * 4;
  float4 o0, o1;
  o0.x = (v0.x - mean) * inv * g[c0 + 0] + b[c0 + 0];
  o0.y = (v0.y - mean) * inv * g[c0 + 1] + b[c0 + 1];
  o0.z = (v0.z - mean) * inv * g[c0 + 2] + b[c0 + 2];
  o0.w = (v0.w - mean) * inv * g[c0 + 3] + b[c0 + 3];
  o1.x = (v1.x - mean) * inv * g[c1 + 0] + b[c1 + 0];
  o1.y = (v1.y - mean) * inv * g[c1 + 1] + b[c1 + 1];
  o1.z = (v1.z - mean) * inv * g[c1 + 2] + b[c1 + 2];
  o1.w = (v1.w - mean) * inv * g[c1 + 3] + b[c1 + 3];
  *(float4*)(yr + c0) = o0;
  *(float4*)(yr + c1) = o1;
}

// --------------------------------------------- pack kv in WMMA-B layout (AV)
// kvA[bh][kt<344][nt<2][lane<32] as float2 : B[k=m][n=hd] tiles
__global__ __launch_bounds__(256) void k_pack_av(const float* __restrict__ y,
                                                 float* __restrict__ kvA) {
  int id = blockIdx.x * 256 + threadIdx.x;   // < B*H*344*2*32
  int lane = id & 31;
  int nt = (id >> 5) & 1;
  int kb = id >> 6;
  int kt = kb % KT;
  int bh = kb / KT;
  int b = bh >> 3, h = bh & 7;
  int hd = 16 * nt + (lane & 15);
  float v[2];
#pragma unroll
  for (int j = 0; j < 2; j++) {
    int m = 4 * kt + 2 * (lane >> 4) + j;
    v[j] = (m < MM)
               ? y[((size_t)(b * NN + 3 * m)) * CC + h * HD + hd]
               : 0.0f;
  }
  ((float2*)kvA)[id] = make_float2(v[0], v[1]);
}

// ----------------------------------------------- pack MLP weights, B layout
__global__ __launch_bounds__(256) void k_pack_w1(const float* __restrict__ w,
                                                 float* __restrict__ wp) {
  int id = blockIdx.x * 256 + threadIdx.x;   // 32*64*32
  int lane = id & 31;
  int ks = (id >> 5) & 63;
  int nt = id >> 11;
  int n = 16 * nt + (lane & 15);
  int k = 4 * ks + 2 * (lane >> 4);
  ((float2*)wp)[id] = make_float2(w[(size_t)k * HID + n],
                                  w[(size_t)(k + 1) * HID + n]);
}

__global__ __launch_bounds__(256) void k_pack_w2(const float* __restrict__ w,
                                                 float* __restrict__ wp) {
  int id = blockIdx.x * 256 + threadIdx.x;   // 16*128*32
  int lane = id & 31;
  int ks = (id >> 5) & 127;
  int nt = id >> 12;
  int n = 16 * nt + (lane & 15);
  int k = 4 * ks + 2 * (lane >> 4);
  ((float2*)wp)[id] = make_float2(w[(size_t)k * CC + n],
                                  w[(size_t)(k + 1) * CC + n]);
}

// ------------------------------------- fused scores -> softmax -> AV -> resid
// one wave per (b, h, 16-query tile); score tile lives in 88 KB LDS.
// kv tiles for QK^T are TDM-streamed (double buffered) into LDS.
__global__ __launch_bounds__(32) void k_attn(const float* __restrict__ x,
                                             const float* __restrict__ y,
                                             const float* __restrict__ kvA,
                                             float* __restrict__ xa,
                                             float* __restrict__ wout) {
  __shared__ float sc[16 * MP];        // 16 x 1376 fp32 = 88064 B
  __shared__ float kvt[2][16 * HD];    // double-buffered TDM kv tile (2x2 KB)
  int lane = threadIdx.x;
  int qt = blockIdx.x & 255;
  int bh = blockIdx.x >> 8;
  int b = bh >> 3, h = bh & 7;
  int n0 = qt * 16;
  int l15 = lane & 15, lh = lane >> 4;
  const float scale = 0.17677669529663687f;   // 1/sqrt(32)

  // Q tile into A-matrix registers: qa[s] covers K = 4s..4s+3
  v2f qa[8];
  const float* qbase = y + ((size_t)(b * NN + n0 + l15)) * CC + h * HD + 2 * lh;
#pragma unroll
  for (int s = 0; s < 8; s++) qa[s] = *(const v2f*)(qbase + 4 * s);

  // TDM-stream kv tiles: rows m = 16t..16t+15 at global rows 3m, cols h*HD..+32
  unsigned kvt_base = (unsigned)(uintptr_t)&kvt[0][0];
  const float* kvsrc = y + ((size_t)b * NN) * CC + h * HD;  // + m*3*CC per row
  tdm_load_kv_tile(kvt_base, kvsrc, MM);

  for (int t = 0; t < MT; t++) {
    if (t + 1 < MT) {
      tdm_load_kv_tile(kvt_base + (unsigned)(((t + 1) & 1) * sizeof(kvt[0])),
                       kvsrc + (size_t)(t + 1) * 16 * 3 * CC,
                       MM - 16 * (t + 1));
      __builtin_amdgcn_s_wait_tensorcnt(1);   // previous tile landed
    } else {
      __builtin_amdgcn_s_wait_tensorcnt(0);   // last tile landed
    }
    asm volatile("" ::: "memory");
    // B operand straight from the LDS tile: kv[key=l15][hd=4s+2lh .. +1]
    const float* kb = &kvt[t & 1][0] + l15 * HD + 2 * lh;
    v8f c = {};
#pragma unroll
    for (int s = 0; s < 8; s++) c = wmma_f32(qa[s], *(const v2f*)(kb + 4 * s), c);
#pragma unroll
    for (int r = 0; r < 8; r++)
      sc[(r + 8 * lh) * MP + t * 16 + l15] = c[r] * scale;
  }

  // row softmax in LDS, streaming normalized probs to the w output
  float* wrow = wout + ((size_t)bh * NN + n0) * MM;
  for (int r = 0; r < 16; r++) {
    float* srow = sc + r * MP;
    float mx = -3.402823e38f;
    for (int i = lane; i < MM; i += 32) mx = fmaxf(mx, srow[i]);
#pragma unroll
    for (int m = 16; m; m >>= 1) mx = fmaxf(mx, __shfl_xor(mx, m));
    float sum = 0.0f;
    for (int i = lane; i < MP; i += 32) {
      float e = 0.0f;
      if (i < MM) {
        e = __expf(srow[i] - mx);
        sum += e;
      }
      srow[i] = e;                       // pad columns forced to 0 for AV
    }
#pragma unroll
    for (int m = 16; m; m >>= 1) sum += __shfl_xor(sum, m);
    float inv = 1.0f / sum;
    for (int i = lane; i < MM; i += 32) {
      float v = srow[i] * inv;
      srow[i] = v;
      wrow[(size_t)r * MM + i] = v;
    }
  }

  // AV: A = prob tile (LDS), B = packed kv stream; two N-tiles cover HD=32
  v8f a0 = {}, a1 = {};
  const v2f* pKA = (const v2f*)kvA + (size_t)bh * (KT * 2 * 32) + lane;
  const float* arow = sc + l15 * MP + 2 * lh;
  for (int kt = 0; kt < KT; kt++) {
    if (kt + 1 < KT)
      __builtin_prefetch((const void*)(pKA + (kt + 1) * 64), 0, 3);
    v2f aw = *(const v2f*)(arow + 4 * kt);
    a0 = wmma_f32(aw, pKA[(kt * 2 + 0) * 32], a0);
    a1 = wmma_f32(aw, pKA[(kt * 2 + 1) * 32], a1);
  }

  // residual: xa = x + attn
#pragma unroll
  for (int r = 0; r < 8; r++) {
    int row = n0 + r + 8 * lh;
    size_t base = ((size_t)(b * NN + row)) * CC + h * HD;
    xa[base + l15] = x[base + l15] + a0[r];
    xa[base + 16 + l15] = x[base + 16 + l15] + a1[r];
  }
}

// ----------------------------- LN2 -> fc1 -> gelu -> fc2 -> residual (fused)
__global__ __launch_bounds__(32) void k_mlp(const float* __restrict__ xa,
                                            const float* __restrict__ g2,
                                            const float* __restrict__ b2,
                                            const float* __restrict__ w1p,
                                            const float* __restrict__ b1,
                                            const float* __restrict__ w2p,
                                            const float* __restrict__ bb2,
                                            float* __restrict__ out1) {
  __shared__ float t16[16 * CC];    // LN2 rows  (16 KB)
  __shared__ float h16[16 * HID];   // hidden    (32 KB)
  int lane = threadIdx.x;
  int l15 = lane & 15, lh = lane >> 4;
  size_t gr0 = (size_t)blockIdx.x * 16;

  // LayerNorm2: lane handles half (lh) of row l15
  {
    const float* xr = xa + (gr0 + l15) * CC + lh * 128;
    float s = 0.0f, q = 0.0f;
#pragma unroll
    for (int i = 0; i < 32; i++) {
      float4 v = *(const float4*)(xr + 4 * i);
      s += v.x + v.y + v.z + v.w;
      q += v.x * v.x + v.y * v.y + v.z * v.z + v.w * v.w;
    }
    s += __shfl_xor(s, 16);
    q += __shfl_xor(q, 16);
    float mean = s * (1.0f / CC);
    float inv = rsqrtf(q * (1.0f / CC) - mean * mean + EPSL);
    float* trow = t16 + l15 * CC + lh * 128;
    const float* gg = g2 + lh * 128;
    const float* bv = b2 + lh * 128;
    for (int i = 0; i < 128; i++)
      trow[i] = (xr[i] - mean) * inv * gg[i] + bv[i];
  }

  // fc1 (256 -> 512) + exact GELU into LDS
  const v2f* pW1 = (const v2f*)w1p + lane;
  const float* arow = t16 + l15 * CC + 2 * lh;
  for (int nt = 0; nt < 32; nt++) {
    v8f acc = {};
    for (int ks = 0; ks < 64; ks++) {
      v2f a = *(const v2f*)(arow + 4 * ks);
      acc = wmma_f32(a, pW1[(nt * 64 + ks) * 32], acc);
    }
    int col = nt * 16 + l15;
    float bias = b1[col];
#pragma unroll
    for (int r = 0; r < 8; r++) {
      float v = acc[r] + bias;
      v = 0.5f * v * (1.0f + erff(v * 0.70710678118654752f));
      h16[(r + 8 * lh) * HID + col] = v;
    }
  }

  // fc2 (512 -> 256) + residual
  const v2f* pW2 = (const v2f*)w2p + lane;
  const float* hrow = h16 + l15 * HID + 2 * lh;
  for (int nt = 0; nt < 16; nt++) {
    v8f acc = {};
    for (int ks = 0; ks < 128; ks++) {
      v2f a = *(const v2f*)(hrow + 4 * ks);
      acc = wmma_f32(a, pW2[(nt * 128 + ks) * 32], acc);
    }
    int col = nt * 16 + l15;
    float bias = bb2[col];
#pragma unroll
    for (int r = 0; r < 8; r++) {
      size_t idx = (gr0 + r + 8 * lh) * CC + col;
      out1[idx] = xa[idx] + acc[r] + bias;
    }
  }
}

// ---------------------------------------------------------------------------
extern "C" void kernel_launch(void* const* d_in, const int* in_sizes, int n_in,
                              void* d_out, int out_size, void* d_ws,
                              size_t ws_size, hipStream_t stream) {
  const float* x     = (const float*)d_in[0];
  const float* ln1_g = (const float*)d_in[1];
  const float* ln1_b = (const float*)d_in[2];
  const float* ln2_g = (const float*)d_in[3];
  const float* ln2_b = (const float*)d_in[4];
  const float* fc1_w = (const float*)d_in[5];
  const float* fc1_b = (const float*)d_in[6];
  const float* fc2_w = (const float*)d_in[7];
  const float* fc2_b = (const float*)d_in[8];

  float* out1 = (float*)d_out;                       // (B,N,C)
  float* wout = out1 + (size_t)BB * NN * CC;         // (B,H,N,M)

  float* ws  = (float*)d_ws;
  float* y   = ws;                                   // B*N*C
  float* xa  = y + (size_t)BB * NN * CC;             // B*N*C
  float* kvA = xa + (size_t)BB * NN * CC;            // B*H*344*2*64
  float* w1p = kvA + (size_t)BB * HH * KT * 2 * 64;  // 32*64*64
  float* w2p = w1p + (size_t)32 * 64 * 64;           // 16*128*64

  k_ln1<<<BB * NN / 8, 256, 0, stream>>>(x, ln1_g, ln1_b, y);
  k_pack_av<<<BB * HH * KT * 2 * 32 / 256, 256, 0, stream>>>(y, kvA);
  k_pack_w1<<<32 * 64 * 32 / 256, 256, 0, stream>>>(fc1_w, w1p);
  k_pack_w2<<<16 * 128 * 32 / 256, 256, 0, stream>>>(fc2_w, w2p);
  k_attn<<<BB * HH * (NN / 16), 32, 0, stream>>>(x, y, kvA, xa, wout);
  k_mlp<<<BB * NN / 16, 32, 0, stream>>>(xa, ln2_g, ln2_b, w1p, fc1_b, w2p,
                                         fc2_b, out1);
}